// RoIPooling_80109730005433
// MI455X (gfx1250) — compile-verified
//
#include <hip/hip_runtime.h>
#include <stdint.h>

#define AS1 __attribute__((address_space(1)))
#define AS3 __attribute__((address_space(3)))

typedef unsigned int uint32x4 __attribute__((ext_vector_type(4)));
typedef int          int32x8  __attribute__((ext_vector_type(8)));
typedef int          int32x4  __attribute__((ext_vector_type(4)));

namespace {
constexpr int   kC       = 256;
constexpr int   kH       = 38;
constexpr int   kW       = 38;
constexpr int   kNRois   = 512;
constexpr int   kPh      = 7;
constexpr int   kPw      = 7;
constexpr int   kBins    = kPh * kPw;          // 49
constexpr int   kChunk   = 4;                  // channels per TDM tile
constexpr int   kBufElts = kChunk * kH * kW;   // 5776 floats per buffer
constexpr int   kThreads = 256;                // 8 wave32s
constexpr float kScale   = 0.0625f;
}

#if __has_builtin(__builtin_amdgcn_tensor_load_to_lds) && __has_builtin(__builtin_amdgcn_s_wait_tensorcnt)
#define HAVE_TDM 1
#else
#define HAVE_TDM 0
#endif

#if HAVE_TDM
// ---- CDNA5 Tensor Data Mover: DMA a 3D tile (x=ww, y=hh rows @ strideY, z=chans @ strideZ)
// from global memory into contiguous LDS. Descriptor per CDNA5 ISA. ----
__device__ __forceinline__ void tdm_load_tile_3d(const float* gbase, float* ldst,
                                                 int tx, int ty, int tz,
                                                 int strideY, int strideZ) {
  const uint64_t ga = (uint64_t)(uintptr_t)gbase;
  const uint32_t la = (uint32_t)(uintptr_t)(AS3 float*)ldst;

  // Group 0 (128b): count=1 (valid user D#), lds_addr[63:32], global_addr[120:64], type[127:126]=2
  uint32x4 g0;
  g0.x = 1u;
  g0.y = la;
  g0.z = (uint32_t)(ga & 0xffffffffu);
  g0.w = (uint32_t)((ga >> 32) & 0x01ffffffu) | (2u << 30);

  // Group 1 (256b)
  const uint32_t td0 = (uint32_t)tx;   // tensor_dim0 == tile_dim0 -> never OOB
  const uint32_t td1 = (uint32_t)ty;   // tensor_dim1 == tile_dim1
  const uint32_t sY  = (uint32_t)strideY;   // tensor_dim0_stride (row stride, elems)
  const uint32_t sZ  = (uint32_t)strideZ;   // tensor_dim1_stride (channel stride, elems)
  int32x8 g1;
  g1[0] = (int)(2u << 16);                                 // mask=0, data_size=2 (4 bytes)
  g1[1] = (int)((td0 & 0xffffu) << 16);                    // abar_addr=0 | tensor_dim0[15:0]
  g1[2] = (int)((td0 >> 16) | ((td1 & 0xffffu) << 16));    // tensor_dim0[31:16] | tensor_dim1[15:0]
  g1[3] = (int)((td1 >> 16) | (((uint32_t)tx) << 16));     // tensor_dim1[31:16] | tile_dim0
  g1[4] = (int)(((uint32_t)ty & 0xffffu) | (((uint32_t)tz) << 16)); // tile_dim1 | tile_dim2
  g1[5] = (int)sY;                                         // tensor_dim0_stride[31:0]
  g1[6] = (int)((sZ & 0xffffu) << 16);                     // dim0_stride[47:32]=0 | dim1_stride[15:0]
  g1[7] = (int)(sZ >> 16);                                 // tensor_dim1_stride[47:16]

  // Group 2 (128b): tensor_dim2 = tz; higher dims unused (tile_dim3=0)
  int32x4 g2;
  g2.x = tz;          // tensor_dim2
  g2.y = 1;           // tensor_dim3 (unused)
  g2.z = 0;           // tensor_dim2_stride[31:0]
  g2.w = 0;           // tensor_dim2_stride[47:32] | tile_dim3=0
  // Group 3 (128b): tensor_dim3_stride=0, tensor_dim4=1, tile_dim4=0
  int32x4 g3;
  g3.x = 0;
  g3.y = (int)(1u << 16);   // dim3_stride[47:32]=0 | tensor_dim4[15:0]=1
  g3.z = 0;                 // tensor_dim4[31:16]=0 | tile_dim4=0
  g3.w = 0;

  // 6-arg toolchain form: 5th slot corresponds to VADDR4 ("unused, set to NULL") -> zeros
  int32x8 g4 = {0, 0, 0, 0, 0, 0, 0, 0};

  __builtin_amdgcn_tensor_load_to_lds(g0, g1, g2, g3, g4, /*cpol=*/0);
}
#else
// ---- fallback: per-lane async global->LDS ----
__device__ __forceinline__ void async_load_f32_to_lds(const float* g, float* l) {
  int* gi = (int*)const_cast<float*>(g);
  int* li = (int*)l;
  __builtin_amdgcn_global_load_async_to_lds_b32((AS1 int*)gi, (AS3 int*)li, 0, 0);
}
#endif

__device__ __forceinline__ int iclamp(int v, int lo, int hi) {
  return v < lo ? lo : (v > hi ? hi : v);
}

__global__ __launch_bounds__(kThreads)
void roi_pool_kernel(const float* __restrict__ feat,
                     const float* __restrict__ rois,
                     float* __restrict__ out) {
  __shared__ float lds[2 * kBufElts];   // double buffer: 46.2 KB

  const int n   = blockIdx.x;           // one ROI per block
  const int tid = threadIdx.x;

  // --- decode ROI (uniform; scalarized) ---
  const float* r = rois + n * 5;
  const int bidx = (int)r[0];
  const int x1 = iclamp((int)(r[1] * kScale), 0, kW - 1);
  const int y1 = iclamp((int)(r[2] * kScale), 0, kH - 1);
  const int x2 = iclamp((int)(r[3] * kScale), 0, kW - 1);
  const int y2 = iclamp((int)(r[4] * kScale), 0, kH - 1);
  const int hh = y2 - y1 + 1;
  const int ww = x2 - x1 + 1;
  const int region = hh * ww;

  const int featBase = ((bidx * kC) * kH) * kW + y1 * kW + x1;  // tile origin, chunk 0
  constexpr int nChunks = kC / kChunk;   // 64

#if HAVE_TDM
  const bool leader = (tid < 32);        // wave 0 drives the DMA engine
  if (leader)
    tdm_load_tile_3d(feat + featBase, &lds[0], ww, hh, kChunk, kW, kH * kW);
#endif

  for (int ci = 0; ci < nChunks; ++ci) {
    const int bufCur = ci & 1;

#if HAVE_TDM
    if (leader) {
      if (ci + 1 < nChunks) {
        // prefetch next chunk into the other buffer while we pool this one
        tdm_load_tile_3d(feat + featBase + (ci + 1) * kChunk * kH * kW,
                         &lds[(bufCur ^ 1) * kBufElts], ww, hh, kChunk, kW, kH * kW);
        __builtin_amdgcn_s_wait_tensorcnt(1);   // in-order => chunk ci complete
      } else {
        __builtin_amdgcn_s_wait_tensorcnt(0);
      }
    }
    __syncthreads();                            // LDS for chunk ci visible to all waves
#else
    {
      const int total = kChunk * region;
      for (int e = tid; e < total; e += kThreads) {
        const int cl  = e / region;
        const int rem = e - cl * region;
        const int ry  = rem / ww;
        const int rx  = rem - ry * ww;
        const float* g = feat + featBase + ((ci * kChunk + cl) * kH + ry) * kW + rx;
        async_load_f32_to_lds(g, &lds[bufCur * kBufElts + e]);
      }
      __builtin_amdgcn_s_wait_asynccnt(0);
      __syncthreads();
    }
#endif

    // ---- adaptive 7x7 max pool from LDS (4 channels * 49 bins = 196 threads) ----
    if (tid < kChunk * kBins) {
      const int cl  = tid / kBins;
      const int bin = tid - cl * kBins;
      const int ph  = bin / kPw;
      const int pw  = bin - ph * kPw;
      const int ys = (ph * hh) / kPh;
      const int ye = ((ph + 1) * hh + kPh - 1) / kPh;
      const int xs = (pw * ww) / kPw;
      const int xe = ((pw + 1) * ww + kPw - 1) / kPw;

      float m = -3.402823466e+38f;   // bins are never empty
      const float* base = &lds[bufCur * kBufElts + cl * region];
      for (int y = ys; y < ye; ++y) {
        const float* row = base + y * ww;
        for (int x = xs; x < xe; ++x)
          m = fmaxf(m, row[x]);
      }
      out[(n * kC + (ci * kChunk + cl)) * kBins + bin] = m;
    }
    __syncthreads();   // all reads of buf done before it is re-targeted by DMA at ci+1
  }
}

extern "C" void kernel_launch(void* const* d_in, const int* in_sizes, int n_in,
                              void* d_out, int out_size, void* d_ws, size_t ws_size,
                              hipStream_t stream) {
  const float* feat = (const float*)d_in[0];   // [4,256,38,38] f32
  const float* rois = (const float*)d_in[1];   // [512,5] f32
  float* out = (float*)d_out;                  // [512,256,7,7] f32

  dim3 grid(kNRois);
  dim3 block(kThreads);
  roi_pool_kernel<<<grid, block, 0, stream>>>(feat, rois, out);
}